// ObstacleCollisionReward_34651796144493
// MI455X (gfx1250) — compile-verified
//
#include <hip/hip_runtime.h>
#include <math.h>

typedef float v2f __attribute__((ext_vector_type(2)));
typedef float v8f __attribute__((ext_vector_type(8)));

#define T_FULL   54
#define NUM_HIST 4
#define T_OUT    50
#define NCHUNK   13   // ceil(50/4)
#define N_AG     32
#define BPG      128
#define FAR_D    1000.0f

#if defined(__has_builtin)
#if __has_builtin(__builtin_amdgcn_wmma_f32_16x16x4_f32)
#define USE_WMMA_F32X4 1
#endif
#endif

__device__ __forceinline__ float u32_as_f(unsigned u) { union { unsigned u; float f; } v; v.u = u; return v.f; }
__device__ __forceinline__ unsigned f_as_u32(float f)  { union { unsigned u; float f; } v; v.f = f; return v.u; }
__device__ __forceinline__ unsigned long long umin64(unsigned long long a, unsigned long long b) { return a < b ? a : b; }

// One wave32 per block. Block handles ego n, timesteps [t0, t0+nt), i.e. a 16-corner
// M-tile (4 t x 4 corners). Loops over 8 N-tiles of 16 boxes each (128 boxes/batch).
__global__ __launch_bounds__(32)
void ObstacleCollision_kernel(const float* __restrict__ infer_position,  // (32,54,2)
                              const float* __restrict__ infer_heading,   // (32,54)
                              const float* __restrict__ box,             // (32,2)
                              const float* __restrict__ soa_position,    // (4096,2)
                              const float* __restrict__ soa_heading,     // (4096,)
                              const float* __restrict__ soa_theta,       // (4096,)
                              const float* __restrict__ soa_length,      // (4096,)
                              float* __restrict__ out_done)              // (32,50) as float
{
    const int lane  = threadIdx.x;          // 0..31
    const int n     = blockIdx.x / NCHUNK;
    const int chunk = blockIdx.x % NCHUNK;
    const int t0    = chunk * 4;
    const int nt    = min(4, T_OUT - t0);   // 4 or 2 (last chunk)

    // Corner this lane materializes: m = lane&15 -> (tl, c)
    const int m   = lane & 15;
    const int tl  = m >> 2;
    const int cc  = m & 3;
    const int tle = min(tl, nt - 1);        // clamp pad corners onto valid data
    const int t   = t0 + tle;

    // Corner world position (matches ref: corners = pos + R(yaw) * (signs*box))
    const int   pidx = n * T_FULL + NUM_HIST + t;
    const float px   = infer_position[2 * pidx + 0];
    const float py   = infer_position[2 * pidx + 1];
    const float yaw  = infer_heading[pidx];
    const float bdx  = box[2 * n + 0];
    const float bdy  = box[2 * n + 1];
    // CORNER_SIGNS = [[.5,.5],[.5,-.5],[-.5,-.5],[-.5,.5]]
    const float sxs = (cc <= 1) ? 0.5f : -0.5f;
    const float sys = (cc == 0 || cc == 3) ? 0.5f : -0.5f;
    const float ox  = sxs * bdx, oy = sys * bdy;
    const float cyw = cosf(yaw), syw = sinf(yaw);
    const float cx  = px + cyw * ox - syw * oy;
    const float cyp = py + syw * ox + cyw * oy;

    // A-matrix 16x4 f32 (2 VGPRs): VGPR0 = K0 | K2 by half-wave, VGPR1 = K1 | K3.
    // Row M = (cx[M], cy[M], 1, 0) so that A.B = ch*cx + sh*cy + tx.
    v2f A;
    A.x = (lane < 16) ? cx  : 1.0f;
    A.y = (lane < 16) ? cyp : 0.0f;

    // Running per-element argmin keys: (dist_bits << 32) | local_box_j.
    // dist >= 0 => float bits monotone; low bits give first-index tie-break,
    // exactly matching jnp.argmin over [edges..., nodes...].
    unsigned long long keyE[8], keyN[8];
#pragma unroll
    for (int e = 0; e < 8; ++e) { keyE[e] = ~0ull; keyN[e] = ~0ull; }

    const int base = n * BPG;

    for (int tile = 0; tile < 8; ++tile) {
        const int   j   = tile * 16 + (lane & 15);   // local box 0..127
        const int   gj  = base + j;
        const float bpx = soa_position[2 * gj + 0];
        const float bpy = soa_position[2 * gj + 1];
        const float hd  = soa_heading[gj];
        const float len = soa_length[gj];
        const float ch  = cosf(hd), sh = sinf(hd);
        const float tx  = -(ch * bpx + sh * bpy);
        const float ty  = sh * bpx - ch * bpy;

        v8f lx8, ly8;
#ifdef USE_WMMA_F32X4
        // B-matrix 4x16 f32 (2 VGPRs), same K mapping convention as A.
        v2f Blx, Bly;
        Blx.x = (lane < 16) ? ch  : tx;  Blx.y = (lane < 16) ? sh : 0.0f;
        Bly.x = (lane < 16) ? -sh : ty;  Bly.y = (lane < 16) ? ch : 0.0f;
        v8f zero = {0.f, 0.f, 0.f, 0.f, 0.f, 0.f, 0.f, 0.f};
        lx8 = __builtin_amdgcn_wmma_f32_16x16x4_f32(false, A, false, Blx, (short)0, zero, false, false);
        ly8 = __builtin_amdgcn_wmma_f32_16x16x4_f32(false, A, false, Bly, (short)0, zero, false, false);
#else
        // Fallback: element (e,lane) = (M = e + 8*(lane>>4), N = lane&15)
        const int hb = (lane >> 4) << 3;
#pragma unroll
        for (int e = 0; e < 8; ++e) {
            const int   M   = e + hb;
            const float cxm = __shfl(cx, M, 32);
            const float cym = __shfl(cyp, M, 32);
            lx8[e] = fmaf(cxm, ch, fmaf(cym, sh, tx));
            ly8[e] = fmaf(cxm, -sh, fmaf(cym, ch, ty));
        }
#endif
        // Elementwise candidates; this lane's box (N = lane&15) is exactly j.
#pragma unroll
        for (int e = 0; e < 8; ++e) {
            const float lxv = lx8[e], lyv = ly8[e];
            const float ed  = (lxv > 0.0f && lxv < len) ? fabsf(lyv) : FAR_D;
            const float nd  = sqrtf(lxv * lxv + lyv * lyv);
            const unsigned long long ek =
                ((unsigned long long)f_as_u32(ed) << 32) | (unsigned)j;
            const unsigned long long nk =
                ((unsigned long long)f_as_u32(nd) << 32) | (unsigned)j;
            keyE[e] = umin64(keyE[e], ek);
            keyN[e] = umin64(keyN[e], nk);
        }
    }

    // Min over the 16 lanes of each half-wave (boxes N=0..15 of all tiles live there).
#pragma unroll
    for (int e = 0; e < 8; ++e) {
#pragma unroll
        for (int off = 1; off < 16; off <<= 1) {
            unsigned long long oe = __shfl_xor(keyE[e], off, 32);
            unsigned long long on = __shfl_xor(keyN[e], off, 32);
            keyE[e] = umin64(keyE[e], oe);
            keyN[e] = umin64(keyN[e], on);
        }
    }

    // Hand winners to one lane per corner: corner M = e + 8*half.
    __shared__ unsigned long long sE[16];
    __shared__ unsigned long long sN[16];
    if ((lane & 15) == 0) {
        const int b0 = (lane >> 4) * 8;
#pragma unroll
        for (int e = 0; e < 8; ++e) { sE[b0 + e] = keyE[e]; sN[b0 + e] = keyN[e]; }
    }
    __syncthreads();

    bool done_c = false;
    if (lane < 16) {
        const unsigned long long ke = sE[lane];
        const unsigned long long kn = sN[lane];
        const float de = u32_as_f((unsigned)(ke >> 32));
        const float dn = u32_as_f((unsigned)(kn >> 32));
        float loss;
        if (de <= dn) {   // edge block precedes node block in the concat -> wins ties
            const int   gj  = base + (int)(ke & 0xffffffffull);
            const float bpx = soa_position[2 * gj + 0];
            const float bpy = soa_position[2 * gj + 1];
            const float hd  = soa_heading[gj];
            const float ch  = cosf(hd), sh = sinf(hd);
            const float dx  = cx - bpx, dy = cyp - bpy;
            const float ly  = -sh * dx + ch * dy;
            loss = fmaxf(ly, 0.0f);                     // edge_loss (THRESHOLD=0)
        } else {
            const int   gj  = base + (int)(kn & 0xffffffffull);
            const float bpx = soa_position[2 * gj + 0];
            const float bpy = soa_position[2 * gj + 1];
            const float hd  = soa_heading[gj];
            const float th  = soa_theta[gj];
            const float ch  = cosf(hd), sh = sinf(hd);
            const float dx  = cx - bpx, dy = cyp - bpy;
            const float lx  = ch * dx + sh * dy;
            const float ly  = -sh * dx + ch * dy;
            const float TWO_PI = 6.28318530717958647692f;
            float ang = atan2f(ly, lx);
            ang = ang - floorf(ang * (1.0f / TWO_PI)) * TWO_PI;   // mod 2pi in [0,2pi)
            const float sgn = (th - ang > 0.0f) ? 1.0f : -1.0f;
            loss = fmaxf(sgn * dn, 0.0f);               // node_loss
        }
        done_c = (loss > 0.0f);
    }

    const unsigned long long b = __ballot(done_c);      // bit m = corner m's done_c
    if (lane < nt) {
        const float v = ((b >> (4 * lane)) & 0xFull) ? 1.0f : 0.0f;  // any over 4 corners
        out_done[n * T_OUT + t0 + lane] = v;
    }
}

// reward[n] = all points not-done = (no done[n,t] set). valid-mask is always true
// (every ego batch has 128 boxes), so reward_c == !done_c exactly.
__global__ __launch_bounds__(32)
void ObstacleReward_kernel(const float* __restrict__ done, float* __restrict__ out_reward)
{
    const int n = threadIdx.x;
    if (n >= N_AG) return;
    float any = 0.0f;
    for (int t = 0; t < T_OUT; ++t) any = fmaxf(any, done[n * T_OUT + t]);
    out_reward[n] = (any > 0.0f) ? 0.0f : 1.0f;
}

extern "C" void kernel_launch(void* const* d_in, const int* in_sizes, int n_in,
                              void* d_out, int out_size, void* d_ws, size_t ws_size,
                              hipStream_t stream) {
    (void)in_sizes; (void)n_in; (void)out_size; (void)d_ws; (void)ws_size;
    const float* infer_position = (const float*)d_in[0];
    const float* infer_heading  = (const float*)d_in[1];
    const float* box            = (const float*)d_in[2];
    const float* soa_position   = (const float*)d_in[3];
    const float* soa_heading    = (const float*)d_in[4];
    const float* soa_theta      = (const float*)d_in[5];
    const float* soa_length     = (const float*)d_in[6];
    // d_in[7]=ego_batch (arange), d_in[8]=soa_batch (repeat) -- structure is fixed.

    float* out = (float*)d_out;                 // [0,1600) done, [1600,1632) reward
    ObstacleCollision_kernel<<<N_AG * NCHUNK, 32, 0, stream>>>(
        infer_position, infer_heading, box,
        soa_position, soa_heading, soa_theta, soa_length, out);
    ObstacleReward_kernel<<<1, 32, 0, stream>>>(out, out + N_AG * T_OUT);
}